// MemNNLayer_72567767433250
// MI455X (gfx1250) — compile-verified
//
#include <hip/hip_runtime.h>
#include <math.h>

#define HOPS  3
#define VOCAB 50000
#define EMBD  128
#define BS    128
#define STORY 50
#define SENT  20
#define QLEN  20

typedef __attribute__((ext_vector_type(2))) float v2f;
typedef __attribute__((ext_vector_type(8))) float v8f;

// ---------------------------------------------------------------------------
// Kernel 1: embedding gather-sums.
//   Blocks [0, 4*BS*STORY)   : S[k][b][s][:] = sum_t A[k][x[b,s,t]][:]
//   Blocks [4*BS*STORY, +BS) : U[b][:]       = sum_t A[0][q[b,t]][:]
// One wave (32 lanes) per row-sum; each lane owns a float4 slice -> each token
// read is one fully-coalesced 512B row.
// ---------------------------------------------------------------------------
__global__ void gather_sum_kernel(const int* __restrict__ x,
                                  const int* __restrict__ q,
                                  const float* __restrict__ A,
                                  float* __restrict__ S,
                                  float* __restrict__ U) {
    const int gid  = blockIdx.x;
    const int lane = threadIdx.x;           // 0..31
    const int nS   = (HOPS + 1) * BS * STORY;
    float4 acc = make_float4(0.f, 0.f, 0.f, 0.f);
    if (gid < nS) {
        const int k   = gid / (BS * STORY);
        const int rem = gid - k * (BS * STORY);
        const int b   = rem / STORY;
        const int s   = rem - b * STORY;
        const int* xr = x + (b * STORY + s) * SENT;
        const float* tab = A + (size_t)k * VOCAB * EMBD;
        #pragma unroll 4
        for (int t = 0; t < SENT; ++t) {
            const int id = xr[t];
            const float4 v = *(const float4*)(tab + (size_t)id * EMBD + lane * 4);
            acc.x += v.x; acc.y += v.y; acc.z += v.z; acc.w += v.w;
        }
        *(float4*)(S + (size_t)gid * EMBD + lane * 4) = acc;
    } else {
        const int b   = gid - nS;
        const int* qr = q + b * QLEN;
        #pragma unroll 4
        for (int t = 0; t < QLEN; ++t) {
            const int id = qr[t];
            const float4 v = *(const float4*)(A + (size_t)id * EMBD + lane * 4);
            acc.x += v.x; acc.y += v.y; acc.z += v.z; acc.w += v.w;
        }
        *(float4*)(U + (size_t)b * EMBD + lane * 4) = acc;
    }
}

// ---------------------------------------------------------------------------
// Kernel 2: all 3 attention hops for one batch element per 128-thread block.
// u kept in LDS; scores by threads 0..49; softmax stats recomputed redundantly
// in registers (50 LDS broadcasts per thread, negligible).
// ---------------------------------------------------------------------------
__global__ void hops_kernel(const float* __restrict__ S,
                            const float* __restrict__ TA,
                            const float* __restrict__ TC,
                            float* __restrict__ U) {
    __shared__ float u_l[EMBD];
    __shared__ float sc[STORY];
    const int b = blockIdx.x;
    const int t = threadIdx.x;              // 0..127
    u_l[t] = U[(size_t)b * EMBD + t];
    __syncthreads();

    for (int k = 0; k < HOPS; ++k) {
        // --- raw attention scores: m . u ---
        if (t < STORY) {
            const float* m  = S + (((size_t)k * BS + b) * STORY + t) * EMBD;
            const float* ta = TA + t * EMBD;
            float acc = 0.f;
            #pragma unroll 8
            for (int e = 0; e < EMBD; ++e)
                acc += (m[e] + ta[e]) * u_l[e];
            sc[t] = acc;
        }
        __syncthreads();

        // --- softmax stats (redundant per-thread, all from LDS broadcasts) ---
        float mx = -1e30f;
        for (int s = 0; s < STORY; ++s) mx = fmaxf(mx, sc[s]);
        float sm = 0.f;
        for (int s = 0; s < STORY; ++s) sm += expf(sc[s] - mx);
        const float inv = 1.0f / sm;

        // --- u += sum_s p[s] * c[s][:]  (c from next table's sums + TC) ---
        const float* cb = S + (((size_t)(k + 1) * BS + b) * STORY) * EMBD + t;
        float unew = u_l[t];
        for (int s = 0; s < STORY; ++s) {
            const float w = expf(sc[s] - mx) * inv;
            unew += w * (cb[(size_t)s * EMBD] + TC[s * EMBD + t]);
        }
        __syncthreads();
        u_l[t] = unew;
        __syncthreads();
    }
    U[(size_t)b * EMBD + t] = u_l[t];
}

// ---------------------------------------------------------------------------
// Kernel 3: out[128,50000] = U[128,128] . A3[50000,128]^T  (exact f32 WMMA).
// One wave owns NT=2 vocab tiles x all 8 row-tiles (full M=128): per k-step it
// issues 2 B loads + 8 A loads into DISTINCT registers, then 16 WMMAs.
// Needs ~160 VGPRs (128 accum + fragments); __launch_bounds__(256, 1) requests
// min-1-wave/EU occupancy so the allocator can use >128 VGPRs without the
// scratch spills seen at the default budget. A3 streamed exactly once.
//
// Fragment addressing (ISA 7.12.2, f32 16x16x4):
//   A: lane<16 -> M=lane,    K={k0,k0+1}; lane>=16 -> M=lane-16, K={k0+2,k0+3}
//   B: transposed analog -> each lane loads float2 of A3 row n at k0+2*khalf
//   D: VGPR r -> M = r + 8*khalf, N = n0 + (lane&15)
// ---------------------------------------------------------------------------
#define NTILES (VOCAB / 16)   // 3125
#define NT 2                  // vocab tiles per wave

__global__ void __launch_bounds__(256, 1)
proj_wmma_kernel(const float* __restrict__ U,
                 const float* __restrict__ A3,
                 float* __restrict__ out) {
    const int lane = threadIdx.x & 31;
    const int wave = threadIdx.x >> 5;
    const int pair = blockIdx.x * (blockDim.x >> 5) + wave;
    if (pair * NT >= NTILES) return;

    const int nl    = lane & 15;
    const int khalf = lane >> 4;            // 0 or 1

    int ncol[NT];
    const float* brow[NT];
    #pragma unroll
    for (int j = 0; j < NT; ++j) {
        int t = pair * NT + j;
        if (t > NTILES - 1) t = NTILES - 1;   // odd tail: duplicate tile (benign)
        ncol[j] = t * 16 + nl;
        brow[j] = A3 + (size_t)ncol[j] * EMBD;
    }
    const float* arow = U + (size_t)nl * EMBD;   // + mt*16*EMBD + kb

    v8f acc[NT][8];
    #pragma unroll
    for (int j = 0; j < NT; ++j)
        #pragma unroll
        for (int mt = 0; mt < 8; ++mt) acc[j][mt] = (v8f)(0.f);

    #pragma unroll 4
    for (int kk = 0; kk < 32; ++kk) {
        const int kb = kk * 4 + khalf * 2;

        // --- batch all loads of this k-step into distinct registers ---
        v2f bv[NT];
        #pragma unroll
        for (int j = 0; j < NT; ++j)
            bv[j] = *(const v2f*)(brow[j] + kb);

        v2f av[8];
        #pragma unroll
        for (int mt = 0; mt < 8; ++mt)
            av[mt] = *(const v2f*)(arow + (size_t)mt * 16 * EMBD + kb);

        // --- 16 WMMAs consuming the batched fragments ---
        #pragma unroll
        for (int mt = 0; mt < 8; ++mt)
            #pragma unroll
            for (int j = 0; j < NT; ++j)
                acc[j][mt] = __builtin_amdgcn_wmma_f32_16x16x4_f32(
                    false, av[mt], false, bv[j], (short)0, acc[j][mt], false, false);
    }

    #pragma unroll
    for (int j = 0; j < NT; ++j)
        #pragma unroll
        for (int mt = 0; mt < 8; ++mt)
            #pragma unroll
            for (int r = 0; r < 8; ++r) {
                const int row = mt * 16 + r + khalf * 8;
                out[(size_t)row * VOCAB + ncol[j]] = acc[j][mt][r];
            }
}

// ---------------------------------------------------------------------------
extern "C" void kernel_launch(void* const* d_in, const int* in_sizes, int n_in,
                              void* d_out, int out_size, void* d_ws, size_t ws_size,
                              hipStream_t stream) {
    const int*   x  = (const int*)d_in[0];
    const int*   q  = (const int*)d_in[1];
    const float* A  = (const float*)d_in[2];
    const float* TA = (const float*)d_in[3];
    const float* TC = (const float*)d_in[4];
    float* out = (float*)d_out;

    // Workspace layout: S [4*128*50*128] f32, then U [128*128] f32 (~12.7 MiB)
    float* S = (float*)d_ws;
    float* U = S + (size_t)(HOPS + 1) * BS * STORY * EMBD;

    const int nS = (HOPS + 1) * BS * STORY;     // 25600 sentence sums
    gather_sum_kernel<<<nS + BS, 32, 0, stream>>>(x, q, A, S, U);

    hops_kernel<<<BS, EMBD, 0, stream>>>(S, TA, TC, U);

    const float* A3 = A + (size_t)HOPS * VOCAB * EMBD;
    const int wavesPerBlock = 8;                // 256 threads
    const int pairs = (NTILES + NT - 1) / NT;   // 1563 waves
    const int grid  = (pairs + wavesPerBlock - 1) / wavesPerBlock;
    proj_wmma_kernel<<<grid, wavesPerBlock * 32, 0, stream>>>(U, A3, out);
}